// CRF_64759516889822
// MI455X (gfx1250) — compile-verified
//
#include <hip/hip_runtime.h>
#include <hip/hip_bf16.h>
#include <math.h>

#define B_   1024
#define S_   1024
#define T_   16
#define CH_  16
#define LEN_ (S_/CH_)   // 64 steps per chunk

typedef __attribute__((ext_vector_type(2))) float v2f;
typedef __attribute__((ext_vector_type(8))) float v8f;

// ds_swizzle with compile-time pattern (group-of-32 mode)
template<int PAT>
__device__ __forceinline__ float swz(float x) {
  return __int_as_float(__builtin_amdgcn_ds_swizzle(__float_as_int(x), PAT));
}
// lanes 0-15 <-> 16-31 swap: xor=0x10, and=0x1f
#define SWAP16(x)  swz<0x401f>(x)
// full-wave xor patterns: (m<<10)|0x1f
#define XOR16(x)   swz<0x401f>(x)
#define XOR8(x)    swz<0x201f>(x)
#define XOR4(x)    swz<0x101f>(x)
#define XOR2(x)    swz<0x081f>(x)
#define XOR1(x)    swz<0x041f>(x)

// ---------------------------------------------------------------------------
// Kernel 1: per-(batch, chunk) linear operator of the CRF forward recurrence,
// in the probability domain with chained V_WMMA_F32_16X16X4_F32.
// State U[k][n] (WMMA B layout): column n = forward vector started from e_n.
// Step: U <- (E^T x U) row-scaled by exp(em[t]), scale folded into A operand.
// Mask handled as a precomputed 64-bit SALU bitmask (identity step if 0).
// Exact power-of-2 renorm every 8 steps, exponents accumulated in esc.
// ---------------------------------------------------------------------------
__global__ __launch_bounds__(32) void crf_chunk_kernel(
    const float* __restrict__ em, const unsigned char* __restrict__ mask,
    const float* __restrict__ trans, float* __restrict__ Uws, int* __restrict__ escws)
{
  const int wid  = blockIdx.x;
  const int b    = wid / CH_;
  const int c    = wid % CH_;
  const int lane = threadIdx.x & 31;
  const int lo   = lane & 15;
  const bool hi  = (lane & 16) != 0;

  // A = exp(transitions)^T in WMMA A layout, 4 K-chunks: A_cc[m=lo][k]
  v2f A[4];
#pragma unroll
  for (int cc = 0; cc < 4; ++cc) {
    int r0 = 4*cc + (hi ? 2 : 0);
    A[cc].x = __expf(trans[r0*16 + lo]);
    A[cc].y = __expf(trans[(r0+1)*16 + lo]);
  }

  const float* emb = em + (size_t)b * (S_*T_);
  const int base = c * LEN_;

  // Collapse this chunk's 64 mask bytes (0/1) into one 64-bit SALU bitmask.
  const unsigned long long* mk8 =
      (const unsigned long long*)(mask + (size_t)b * S_ + base);
  unsigned long long bm = 0ull;
#pragma unroll
  for (int w = 0; w < 8; ++w) {
    unsigned long long v8 = mk8[w];
    bm |= (((v8 * 0x0102040810204080ull) >> 56) & 0xffull) << (8*w);
  }

  // State in WMMA B layout: Bst[cc].x = U[4cc + (hi?2:0)][lo], .y = next k
  v2f Bst[4];
  int t0;
  if (c == 0) {
    // chunk 0 absorbs t=0: U = diag(exp(em[b,0,:]))
    float f0 = __expf(emb[lo]);
#pragma unroll
    for (int cc = 0; cc < 4; ++cc) {
      int k0 = 4*cc + (hi ? 2 : 0);
      Bst[cc].x = (k0   == lo) ? f0 : 0.f;
      Bst[cc].y = (k0+1 == lo) ? f0 : 0.f;
    }
    t0 = 1;
  } else {
#pragma unroll
    for (int cc = 0; cc < 4; ++cc) {
      int k0 = 4*cc + (hi ? 2 : 0);
      Bst[cc].x = (k0   == lo) ? 1.f : 0.f;
      Bst[cc].y = (k0+1 == lo) ? 1.f : 0.f;
    }
    t0 = base;
  }
  const int tend = base + LEN_;

  // 2-deep rotating prefetch of the per-step emission scale (L2-resident).
  int tp1 = (t0+1 < tend) ? t0+1 : tend-1;
  float e0 = emb[(size_t)t0  * T_ + lo];
  float e1 = emb[(size_t)tp1 * T_ + lo];

  int esc = 0;
  for (int t = t0; t < tend; ++t) {
    int tn = (t+2 < tend) ? t+2 : tend-1;
    float e2 = emb[(size_t)tn * T_ + lo];   // prefetch, no wait yet
    float f = __expf(e0);
    if ((bm >> (unsigned)(t - base)) & 1ull) {   // wave-uniform SALU test
      v2f a0 = A[0]*f, a1 = A[1]*f, a2 = A[2]*f, a3 = A[3]*f;
      v8f d = {};
      d = __builtin_amdgcn_wmma_f32_16x16x4_f32(false, a0, false, Bst[0], (short)0, d, false, false);
      d = __builtin_amdgcn_wmma_f32_16x16x4_f32(false, a1, false, Bst[1], (short)0, d, false, false);
      d = __builtin_amdgcn_wmma_f32_16x16x4_f32(false, a2, false, Bst[2], (short)0, d, false, false);
      d = __builtin_amdgcn_wmma_f32_16x16x4_f32(false, a3, false, Bst[3], (short)0, d, false, false);
      // D-layout -> B-layout feedback: 8 half-wave swaps + 8 selects
      float s0 = SWAP16(d[0]), s1 = SWAP16(d[1]), s2 = SWAP16(d[2]), s3 = SWAP16(d[3]);
      float s4 = SWAP16(d[4]), s5 = SWAP16(d[5]), s6 = SWAP16(d[6]), s7 = SWAP16(d[7]);
      Bst[0].x = hi ? s2   : d[0];  Bst[0].y = hi ? s3   : d[1];
      Bst[1].x = hi ? s6   : d[4];  Bst[1].y = hi ? s7   : d[5];
      Bst[2].x = hi ? d[2] : s0;    Bst[2].y = hi ? d[3] : s1;
      Bst[3].x = hi ? d[6] : s4;    Bst[3].y = hi ? d[7] : s5;
    }
    if ((t & 7) == 7) {   // exact power-of-2 renorm (hits final step too)
      float mx = fmaxf(fmaxf(fmaxf(Bst[0].x, Bst[0].y), fmaxf(Bst[1].x, Bst[1].y)),
                       fmaxf(fmaxf(Bst[2].x, Bst[2].y), fmaxf(Bst[3].x, Bst[3].y)));
      mx = fmaxf(mx, XOR16(mx));
      mx = fmaxf(mx, XOR8(mx));
      mx = fmaxf(mx, XOR4(mx));
      mx = fmaxf(mx, XOR2(mx));
      mx = fmaxf(mx, XOR1(mx));
      int e = ((__float_as_int(mx) >> 23) & 0xff) - 127;
      if (e != 0 && mx > 0.f) {
        float sc = __int_as_float((127 - e) << 23);   // 2^-e exactly
#pragma unroll
        for (int cc = 0; cc < 4; ++cc) { Bst[cc].x *= sc; Bst[cc].y *= sc; }
        esc += e;
      }
    }
    e0 = e1; e1 = e2;
  }

  // Store U[k][n] row-major (rows of U = contiguous columns for combine).
  float* Uo = Uws + ((size_t)b * CH_ + c) * 256;
#pragma unroll
  for (int cc = 0; cc < 4; ++cc) {
    int k0 = 4*cc + (hi ? 2 : 0);
    Uo[k0*16 + lo]     = Bst[cc].x;
    Uo[(k0+1)*16 + lo] = Bst[cc].y;
  }
  if (lane == 0) escws[b*CH_ + c] = esc;
}

// ---------------------------------------------------------------------------
// Kernel 2: fold v = exp(start) through the C chunk operators per batch:
// v'[j] = sum_i v[i] * U[j][i]  (lane j reads its row of U as 4x b128).
// logZ[b] = log(v . exp(end)) + ln2 * accumulated exponent.
// Two batches per wave (16-lane groups).
// ---------------------------------------------------------------------------
__global__ __launch_bounds__(32) void crf_combine_kernel(
    const float* __restrict__ Uws, const int* __restrict__ escws,
    const float* __restrict__ startT, const float* __restrict__ endT,
    float* __restrict__ logZ)
{
  const int lane = threadIdx.x & 31;
  const int j    = lane & 15;
  const int b    = blockIdx.x * 2 + (lane >> 4);

  float v = __expf(startT[j]);
  int E = 0;
  for (int cix = 0; cix < CH_; ++cix) {
    const float4* Ur = (const float4*)(Uws + ((size_t)b * CH_ + cix) * 256 + j*16);
    float4 r0 = Ur[0], r1 = Ur[1], r2 = Ur[2], r3 = Ur[3];
    float nv = 0.f;
    // broadcast v from lane i of this 16-lane group: or_mask=i, and_mask=0x10
#define ACC(I, VAL) nv = fmaf(swz<((I)<<5) | 0x10>(v), (VAL), nv)
    ACC(0,  r0.x); ACC(1,  r0.y); ACC(2,  r0.z); ACC(3,  r0.w);
    ACC(4,  r1.x); ACC(5,  r1.y); ACC(6,  r1.z); ACC(7,  r1.w);
    ACC(8,  r2.x); ACC(9,  r2.y); ACC(10, r2.z); ACC(11, r2.w);
    ACC(12, r3.x); ACC(13, r3.y); ACC(14, r3.z); ACC(15, r3.w);
#undef ACC
    v = nv;
    E += escws[b*CH_ + cix];
    float mx = v;
    mx = fmaxf(mx, XOR8(mx));
    mx = fmaxf(mx, XOR4(mx));
    mx = fmaxf(mx, XOR2(mx));
    mx = fmaxf(mx, XOR1(mx));
    int e = ((__float_as_int(mx) >> 23) & 0xff) - 127;
    if (e != 0 && mx > 0.f) { v *= __int_as_float((127 - e) << 23); E += e; }
  }
  v *= __expf(endT[j]);
  float s = v;
  s += XOR8(s);
  s += XOR4(s);
  s += XOR2(s);
  s += XOR1(s);
  if (j == 0) logZ[b] = __logf(s) + (float)E * 0.69314718055994530942f;
}

// ---------------------------------------------------------------------------
// Kernel 3: gold path score (gathers + block reduction), one block per batch.
// ---------------------------------------------------------------------------
__global__ __launch_bounds__(256) void crf_gold_kernel(
    const float* __restrict__ em, const long long* __restrict__ tags,
    const unsigned char* __restrict__ mask, const float* __restrict__ trans,
    const float* __restrict__ startT, const float* __restrict__ endT,
    float* __restrict__ gold)
{
  const int b = blockIdx.x;
  const int tid = threadIdx.x;
  const long long* tg = tags + (size_t)b * S_;
  const unsigned char* mk = mask + (size_t)b * S_;
  const float* emb = em + (size_t)b * (S_*T_);

  float acc = 0.f;
  int mc = 0;
  for (int t = tid; t < S_; t += 256) mc += mk[t];
  for (int t = tid + 1; t < S_; t += 256) {
    int jt = (int)tg[t], jp = (int)tg[t-1];
    if (mk[t]) acc += emb[(size_t)t*T_ + jt] + trans[jt*16 + jp];  // transitions[cur, prev]
  }
  __shared__ float sacc[256];
  __shared__ int   smc[256];
  sacc[tid] = acc; smc[tid] = mc;
  __syncthreads();
  for (int stp = 128; stp >= 1; stp >>= 1) {
    if (tid < stp) { sacc[tid] += sacc[tid+stp]; smc[tid] += smc[tid+stp]; }
    __syncthreads();
  }
  if (tid == 0) {
    int j0 = (int)tg[0];
    int last = smc[0] - 1;
    int jl = (int)tg[last];
    gold[b] = startT[j0] + emb[j0] + sacc[0] + endT[jl];
  }
}

// ---------------------------------------------------------------------------
// Kernel 4: mean(logZ - gold)
// ---------------------------------------------------------------------------
__global__ __launch_bounds__(1024) void crf_nll_kernel(
    const float* __restrict__ logZ, const float* __restrict__ gold,
    float* __restrict__ out)
{
  __shared__ float sh[1024];
  int tid = threadIdx.x;
  sh[tid] = logZ[tid] - gold[tid];
  __syncthreads();
  for (int stp = 512; stp >= 1; stp >>= 1) {
    if (tid < stp) sh[tid] += sh[tid+stp];
    __syncthreads();
  }
  if (tid == 0) out[0] = sh[0] * (1.f / 1024.f);
}

extern "C" void kernel_launch(void* const* d_in, const int* in_sizes, int n_in,
                              void* d_out, int out_size, void* d_ws, size_t ws_size,
                              hipStream_t stream) {
  const float*         em     = (const float*)d_in[0];
  const long long*     tags   = (const long long*)d_in[1];
  const unsigned char* mask   = (const unsigned char*)d_in[2];
  const float*         trans  = (const float*)d_in[3];
  const float*         startT = (const float*)d_in[4];
  const float*         endT   = (const float*)d_in[5];
  float* out = (float*)d_out;

  float* Uws   = (float*)d_ws;                          // B*CH*256 floats (16 MB)
  int*   escws = (int*)(Uws + (size_t)B_*CH_*256);      // B*CH ints
  float* logZ  = (float*)(escws + B_*CH_);              // B floats
  float* gold  = logZ + B_;                             // B floats

  crf_chunk_kernel<<<B_*CH_, 32, 0, stream>>>(em, mask, trans, Uws, escws);
  crf_combine_kernel<<<B_/2, 32, 0, stream>>>(Uws, escws, startT, endT, logZ);
  crf_gold_kernel<<<B_, 256, 0, stream>>>(em, tags, mask, trans, startT, endT, gold);
  crf_nll_kernel<<<1, 1024, 0, stream>>>(logZ, gold, out);
}